// HashGrid1D_19645180412085
// MI455X (gfx1250) — compile-verified
//
#include <hip/hip_runtime.h>

// HashGrid1D encoding for MI455X (gfx1250, wave32).
//
// Thread mapping: gid = point*16 + level. Lanes 0..15 of a wave handle one
// point's 16 levels -> the 16 float4 stores from those lanes are contiguous
// (256B), fully coalesced; x[point] is one broadcast load per half-wave.
//
// Memory policy:
//   - tables (134 MB) fit in the 192 MB L2 -> gathers use default RT hint.
//   - output (512 MB, write-once, never re-read) uses non-temporal b128
//     stores so the streaming writes do not evict the table from L2.

#define L_LEVELS 16
#define H_SIZE   524288u
#define H_MASK   (H_SIZE - 1u)
#define P1       73856093u
#define P2       19349663u

typedef __attribute__((ext_vector_type(4))) float v4f;

// (res - 1) per level, res = round(logspace(log10(16), log10(1024), 16))
__constant__ float c_resm1[L_LEVELS] = {
    15.f,  20.f,  27.f,  36.f,  48.f,  63.f,  83.f,  110.f,
    146.f, 193.f, 255.f, 337.f, 445.f, 587.f, 775.f, 1023.f
};

__global__ __launch_bounds__(256) void hashgrid1d_kernel(
    const float* __restrict__ X,      // (B,)
    const float* __restrict__ T,      // (L, H, 4) row-major
    float* __restrict__ out,          // (B, 64)
    int B)
{
    int gid   = blockIdx.x * blockDim.x + threadIdx.x;
    int level = gid & (L_LEVELS - 1);
    int point = gid >> 4;
    if (point >= B) return;

    float x = X[point];
    x = fminf(fmaxf(x, 0.0f), 1.0f);

    float rm1 = c_resm1[level];
    float t   = x * rm1;
    float fi  = floorf(t);
    float w   = t - fi;

    int i0 = (int)fi;
    int i1 = min(i0 + 1, (int)rm1);

    // int32 wraparound multiply == uint32 multiply (low bits identical);
    // mod 2^19 of floored-mod == bitwise AND even for negative int32.
    unsigned lvlh = (unsigned)level * P2;
    unsigned h0 = (((unsigned)i0 * P1) ^ lvlh) & H_MASK;
    unsigned h1 = (((unsigned)i1 * P1) ^ lvlh) & H_MASK;

    const v4f* tab = (const v4f*)T + (size_t)level * (size_t)H_SIZE;
    v4f e0 = tab[h0];   // global_load_b128, RT hint (stay in L2)
    v4f e1 = tab[h1];

    float om = 1.0f - w;
    v4f e = e0 * om + e1 * w;   // matches reference e0*(1-w) + e1*w

    // Streaming output: non-temporal 128-bit store keeps tables L2-resident.
    __builtin_nontemporal_store(e, (v4f*)out + gid);
}

extern "C" void kernel_launch(void* const* d_in, const int* in_sizes, int n_in,
                              void* d_out, int out_size, void* d_ws, size_t ws_size,
                              hipStream_t stream)
{
    const float* X = (const float*)d_in[0];   // x: (B,) float32
    const float* T = (const float*)d_in[1];   // tables: (16, 524288, 4) float32
    float* out = (float*)d_out;               // (B, 64) float32

    int B = in_sizes[0];
    long long total = (long long)B * L_LEVELS;
    int block = 256;
    int grid  = (int)((total + block - 1) / block);

    hashgrid1d_kernel<<<grid, block, 0, stream>>>(X, T, out, B);
}